// myModel_64553358458978
// MI455X (gfx1250) — compile-verified
//
#include <hip/hip_runtime.h>
#include <hip/hip_bf16.h>

typedef float v2f __attribute__((ext_vector_type(2)));
typedef float v8f __attribute__((ext_vector_type(8)));

#define Bdim 16
#define Vdim 20000
#define Ddim 256
#define Hdim 256
#define LDA  260   // padded LDS row stride (floats) for WMMA A-tile
#define LDT  257   // padded LDS row stride for ah tile: (lane+h)%64 -> conflict-free
#define EPSV 1e-5f

// ---------------------------------------------------------------------------
// Fused:  T = A_tile(16x256) @ W1g(256x256) + bias1 ; T = LN(T; gamma,beta) ;
//         Out_tile = T @ W2g(256x256) (+ bias2 if non-null)
// One block = 16 rows. 8 waves; wave w owns output columns [w*32, w*32+32)
// as two 16x16 WMMA tiles. f32 WMMA 16x16x4, K stepped by 4.
// ---------------------------------------------------------------------------
__global__ __launch_bounds__(256) void proj_ln_proj_kernel(
    const float* __restrict__ Ain,    // (nrows,256), nrows = gridDim.x*16
    const float* __restrict__ W1g,    // (256,256) row-major
    const float* __restrict__ bias1,  // (256)
    const float* __restrict__ gamma,  // (256)
    const float* __restrict__ beta,   // (256)
    const float* __restrict__ W2g,    // (256,256) row-major
    const float* __restrict__ bias2,  // (256) or nullptr
    float* __restrict__ Out)          // (nrows,256)
{
    __shared__ float Alds[16 * LDA];
    __shared__ float part_s[16][16];
    __shared__ float part_q[16][16];
    __shared__ float mean_s[16];
    __shared__ float rstd_s[16];

    const int t    = threadIdx.x;
    const int lane = t & 31;
    const int wave = t >> 5;
    const int v0   = blockIdx.x * 16;

    // ---- stage A tile (16x256) into LDS, coalesced ----
#pragma unroll
    for (int i = 0; i < 16; ++i) {
        int idx = t + 256 * i;            // 0..4095
        int r = idx >> 8, c = idx & 255;
        Alds[r * LDA + c] = Ain[(v0 + r) * Ddim + c];
    }
    __syncthreads();

    // ---- WMMA fragment coordinates (ISA 7.12.2 layouts) ----
    const int mrow = lane & 15;           // A frag: M = lane%16
    const int kb   = (lane >> 4) << 1;    // A/B frag: K sub-pair 0 or 2
    const int ncol = (wave << 5) + mrow;  // B/C/D frag: N = lane%16 within tile

    v8f acc0 = {}; v8f acc1 = {};
#pragma unroll 8
    for (int k = 0; k < 256; k += 4) {
        v2f a;
        a.x = Alds[mrow * LDA + k + kb];
        a.y = Alds[mrow * LDA + k + kb + 1];
        v2f b0, b1v;
        b0.x  = W1g[(k + kb) * Hdim + ncol];
        b0.y  = W1g[(k + kb + 1) * Hdim + ncol];
        b1v.x = W1g[(k + kb) * Hdim + ncol + 16];
        b1v.y = W1g[(k + kb + 1) * Hdim + ncol + 16];
        acc0 = __builtin_amdgcn_wmma_f32_16x16x4_f32(false, a, false, b0,
                                                     (short)0, acc0, false, false);
        acc1 = __builtin_amdgcn_wmma_f32_16x16x4_f32(false, a, false, b1v,
                                                     (short)0, acc1, false, false);
    }
    __syncthreads();   // all GEMM1 reads of Alds done before overwrite

    // ---- write GEMM1 result (+bias1) back into Alds (C/D layout) ----
    const int crow = (lane >> 4) << 3;        // 0 or 8
    const int ccol = (wave << 5) + (lane & 15);
#pragma unroll
    for (int j = 0; j < 8; ++j) {
        int r = crow + j;
        Alds[r * LDA + ccol]      = acc0[j] + bias1[ccol];
        Alds[r * LDA + ccol + 16] = acc1[j] + bias1[ccol + 16];
    }
    __syncthreads();

    // ---- LayerNorm over each of the 16 rows (256 cols) ----
    {
        int r = t >> 4, j = t & 15;
        float s = 0.f, q = 0.f;
#pragma unroll
        for (int i = 0; i < 16; ++i) {
            float x = Alds[r * LDA + j + 16 * i];
            s += x; q += x * x;
        }
        part_s[r][j] = s; part_q[r][j] = q;
    }
    __syncthreads();
    if (t < 16) {
        float s = 0.f, q = 0.f;
#pragma unroll
        for (int j = 0; j < 16; ++j) { s += part_s[t][j]; q += part_q[t][j]; }
        float m  = s * (1.f / 256.f);
        float vv = q * (1.f / 256.f) - m * m;
        mean_s[t] = m;
        rstd_s[t] = rsqrtf(vv + EPSV);
    }
    __syncthreads();
    {
        int r = t >> 4, j = t & 15;
        float m = mean_s[r], rs = rstd_s[r];
#pragma unroll
        for (int i = 0; i < 16; ++i) {
            int c = j + 16 * i;
            float x = Alds[r * LDA + c];
            Alds[r * LDA + c] = (x - m) * rs * gamma[c] + beta[c];
        }
    }
    __syncthreads();

    // ---- GEMM2: LN(T) @ W2g ----
    acc0 = (v8f){}; acc1 = (v8f){};
#pragma unroll 8
    for (int k = 0; k < 256; k += 4) {
        v2f a;
        a.x = Alds[mrow * LDA + k + kb];
        a.y = Alds[mrow * LDA + k + kb + 1];
        v2f b0, b1v;
        b0.x  = W2g[(k + kb) * Hdim + ncol];
        b0.y  = W2g[(k + kb + 1) * Hdim + ncol];
        b1v.x = W2g[(k + kb) * Hdim + ncol + 16];
        b1v.y = W2g[(k + kb + 1) * Hdim + ncol + 16];
        acc0 = __builtin_amdgcn_wmma_f32_16x16x4_f32(false, a, false, b0,
                                                     (short)0, acc0, false, false);
        acc1 = __builtin_amdgcn_wmma_f32_16x16x4_f32(false, a, false, b1v,
                                                     (short)0, acc1, false, false);
    }

    float bb0 = bias2 ? bias2[ccol]      : 0.f;
    float bb1 = bias2 ? bias2[ccol + 16] : 0.f;
#pragma unroll
    for (int j = 0; j < 8; ++j) {
        int r = crow + j;
        Out[(v0 + r) * Hdim + ccol]      = acc0[j] + bb0;
        Out[(v0 + r) * Hdim + ccol + 16] = acc1[j] + bb1;
    }
}

// ---------------------------------------------------------------------------
// scores[b,v] = sum_h tanh(ph[b,h] + ah[v,h]) * W2[h] + b2   (b1 folded in ph)
// Transposed mapping: lane = vocab row within a 32-row tile (no cross-lane
// reduction, coalesced stores); wave = batch pair (b, b+8) sharing ah/W2 reads.
// ---------------------------------------------------------------------------
__device__ __forceinline__ float fast_tanh(float x) {
    float e = __expf(2.0f * x);       // v_exp_f32 (TRANS32) path
    return 1.0f - 2.0f / (e + 1.0f);
}

__global__ __launch_bounds__(256) void scores_kernel(
    const float* __restrict__ ah,     // (V,256)
    const float* __restrict__ ph,     // (16,256) already includes b1
    const float* __restrict__ W2,     // (256)
    const float* __restrict__ b2,     // (1)
    float* __restrict__ scores)       // (16,V)
{
    __shared__ float phs[16 * 256];     // 16 KB
    __shared__ float w2s[256];          // 1 KB
    __shared__ float aht[32 * LDT];     // ~32.9 KB, bank-conflict-free rows

    const int t  = threadIdx.x;
    const int v0 = blockIdx.x * 32;

    // stage ph (16x256) and W2
#pragma unroll
    for (int i = 0; i < 16; ++i) phs[t + 256 * i] = ph[t + 256 * i];
    w2s[t] = W2[t];
    // stage ah tile (32x256), coalesced: consecutive t -> consecutive cols
#pragma unroll
    for (int i = 0; i < 32; ++i) {
        int idx = t + 256 * i;          // 0..8191
        int r = idx >> 8, c = idx & 255;
        aht[r * LDT + c] = ah[(v0 + r) * 256 + c];
    }
    __syncthreads();

    const int lane = t & 31;            // vocab row within tile
    const int wave = t >> 5;            // batch pair selector
    const int b0 = wave, b1 = wave + 8;
    const float bb2 = b2[0];

    float s0 = 0.f, s1 = 0.f;
#pragma unroll 4
    for (int h = 0; h < 256; ++h) {
        float av = aht[lane * LDT + h];     // per-lane, conflict-free
        float w  = w2s[h];                  // same-address broadcast
        s0 += fast_tanh(phs[b0 * 256 + h] + av) * w;
        s1 += fast_tanh(phs[b1 * 256 + h] + av) * w;
    }
    scores[b0 * Vdim + v0 + lane] = s0 + bb2;   // coalesced 32-wide stores
    scores[b1 * Vdim + v0 + lane] = s1 + bb2;
}

// ---------------------------------------------------------------------------
// Final LayerNorm over the vocab dim: one block per batch row.
// ---------------------------------------------------------------------------
__global__ __launch_bounds__(256) void ln_vocab_kernel(
    const float* __restrict__ scores, // (16,V)
    const float* __restrict__ g,      // (V)
    const float* __restrict__ bv,     // (V)
    float* __restrict__ out)          // (16,V)
{
    __shared__ float rs[256];
    __shared__ float rq[256];
    const int b = blockIdx.x;
    const int t = threadIdx.x;

    float s = 0.f, q = 0.f;
    for (int v = t; v < Vdim; v += 256) {
        float x = scores[b * Vdim + v];
        s += x; q += x * x;
    }
    rs[t] = s; rq[t] = q;
    __syncthreads();
#pragma unroll
    for (int off = 128; off > 0; off >>= 1) {
        if (t < off) { rs[t] += rs[t + off]; rq[t] += rq[t + off]; }
        __syncthreads();
    }
    const float invV = 1.0f / (float)Vdim;
    float m  = rs[0] * invV;
    float vv = rq[0] * invV - m * m;
    float rstd = rsqrtf(vv + EPSV);

    for (int v = t; v < Vdim; v += 256) {
        float x = scores[b * Vdim + v];
        out[b * Vdim + v] = (x - m) * rstd * g[v] + bv[v];
    }
}

// ---------------------------------------------------------------------------
extern "C" void kernel_launch(void* const* d_in, const int* in_sizes, int n_in,
                              void* d_out, int out_size, void* d_ws, size_t ws_size,
                              hipStream_t stream) {
    const float* patient = (const float*)d_in[0];   // (16,256)
    const float* atc4    = (const float*)d_in[1];   // (V,256)
    const float* Wp      = (const float*)d_in[2];
    const float* bp      = (const float*)d_in[3];
    const float* gp      = (const float*)d_in[4];
    const float* betap   = (const float*)d_in[5];
    const float* Wa      = (const float*)d_in[6];
    const float* ba      = (const float*)d_in[7];
    const float* ga      = (const float*)d_in[8];
    const float* betaa   = (const float*)d_in[9];
    const float* W1      = (const float*)d_in[10];  // (512,256)
    const float* b1      = (const float*)d_in[11];
    const float* W2      = (const float*)d_in[12];
    const float* b2      = (const float*)d_in[13];
    const float* g_pred  = (const float*)d_in[14];
    const float* b_pred  = (const float*)d_in[15];

    float* ws     = (float*)d_ws;
    float* ah     = ws;                              // V*256
    float* ph     = ah + (size_t)Vdim * Ddim;        // 16*256 (includes +b1)
    float* scores = ph + (size_t)Bdim * Ddim;        // 16*V

    // patient tower: p = LN(patient@Wp+bp), ph = p@W1[:D] + b1
    proj_ln_proj_kernel<<<1, 256, 0, stream>>>(
        patient, Wp, bp, gp, betap, W1 /* rows 0..255 */, b1, ph);

    // atc4 tower: a = LN(atc4@Wa+ba), ah = a@W1[D:]
    proj_ln_proj_kernel<<<Vdim / 16, 256, 0, stream>>>(
        atc4, Wa, ba, ga, betaa, W1 + (size_t)Ddim * Hdim, nullptr, ah);

    // fused tanh attention scores (32-v tiles, no cross-lane reductions)
    scores_kernel<<<Vdim / 32, 256, 0, stream>>>(ah, ph, W2, b2, scores);

    // layernorm over vocab -> output
    ln_vocab_kernel<<<Bdim, 256, 0, stream>>>(scores, g_pred, b_pred, (float*)d_out);
}